// GNN_90606630076834
// MI455X (gfx1250) — compile-verified
//
#include <hip/hip_runtime.h>
#include <hip/hip_bf16.h>
#include <math.h>

typedef float v2f __attribute__((ext_vector_type(2)));
typedef float v8f __attribute__((ext_vector_type(8)));

#define NODE_DIM 128
#define LDS_STRIDE 132          // 128 + 4 pad floats: kills bank conflicts, keeps 16B align
#define WAVES_PER_BLOCK 2

// ---------------------------------------------------------------------------
// Degree count: one thread per edge, f32 atomic (resolves in L2).
// ---------------------------------------------------------------------------
__global__ void sage_count_kernel(const int* __restrict__ dst,
                                  float* __restrict__ cnt, int E) {
  int e = blockIdx.x * blockDim.x + threadIdx.x;
  if (e < E) atomicAdd(&cnt[dst[e]], 1.0f);
}

// ---------------------------------------------------------------------------
// Edge scatter: 32 lanes per edge, each lane moves a float4 (coalesced 512B
// row gather; source table is L2-resident), 4 f32 atomics into agg.
// ---------------------------------------------------------------------------
__global__ void sage_scatter_kernel(const float* __restrict__ feat,
                                    const int* __restrict__ src,
                                    const int* __restrict__ dst,
                                    float* __restrict__ agg, int E) {
  long long gid = (long long)blockIdx.x * blockDim.x + threadIdx.x;
  int e = (int)(gid >> 5);
  int q = (int)(gid & 31);
  if (e >= E) return;
  int s = src[e];
  int d = dst[e];
  const float4 v = *reinterpret_cast<const float4*>(
      feat + (long long)s * NODE_DIM + q * 4);
  float* base = agg + (long long)d * NODE_DIM + q * 4;
  atomicAdd(base + 0, v.x);
  atomicAdd(base + 1, v.y);
  atomicAdd(base + 2, v.z);
  atomicAdd(base + 3, v.w);
}

// ---------------------------------------------------------------------------
// Combine: out = (agg/deg) @ Wl^T + b + xin @ Wr^T  [+ L2-normalize + ReLU]
// One wave32 per 16-node tile; WMMA f32 16x16x4, K=128 in steps of 4,
// 8 column tiles of 16 outputs each.
//
// Fragment layouts (CDNA5 ISA 7.12.2, wave32):
//   A 16x4 f32 : lane L<16 holds M=L  {V0:K=0, V1:K=1}; L>=16: M=L-16 {K=2,K=3}
//   B 4x16 f32 : lane L<16 holds N=L  {V0:K=0, V1:K=1}; L>=16: N=L-16 {K=2,K=3}
//   C/D 16x16  : VGPR v, lanes 0-15: M=v,N=lane; lanes 16-31: M=v+8,N=lane-16
// ---------------------------------------------------------------------------
template <bool NORM_RELU>
__global__ __launch_bounds__(WAVES_PER_BLOCK * 32)
void sage_combine_kernel(const float* __restrict__ agg,
                         const float* __restrict__ cnt,
                         const float* __restrict__ xin,
                         const float* __restrict__ Wl,
                         const float* __restrict__ bias,
                         const float* __restrict__ Wr,
                         float* __restrict__ outp, int N) {
  __shared__ float s_agg[WAVES_PER_BLOCK][16][LDS_STRIDE];
  __shared__ float s_in [WAVES_PER_BLOCK][16][LDS_STRIDE];
  __shared__ float s_out[WAVES_PER_BLOCK][16][LDS_STRIDE];

  const int lane  = threadIdx.x & 31;
  const int wv    = threadIdx.x >> 5;
  const int tile  = blockIdx.x * WAVES_PER_BLOCK + wv;
  const int node0 = tile * 16;

  // ---- Stage A-tiles into LDS (coalesced float4 per lane; mean applied) ----
  #pragma unroll 4
  for (int r = 0; r < 16; ++r) {
    const int n = node0 + r;
    if (n < N) {                       // uniform across the wave per iteration
      const float rc = 1.0f / fmaxf(cnt[n], 1.0f);
      const float4 va = *reinterpret_cast<const float4*>(
          agg + (long long)n * NODE_DIM + lane * 4);
      const float4 vx = *reinterpret_cast<const float4*>(
          xin + (long long)n * NODE_DIM + lane * 4);
      s_agg[wv][r][lane * 4 + 0] = va.x * rc;
      s_agg[wv][r][lane * 4 + 1] = va.y * rc;
      s_agg[wv][r][lane * 4 + 2] = va.z * rc;
      s_agg[wv][r][lane * 4 + 3] = va.w * rc;
      s_in[wv][r][lane * 4 + 0] = vx.x;
      s_in[wv][r][lane * 4 + 1] = vx.y;
      s_in[wv][r][lane * 4 + 2] = vx.z;
      s_in[wv][r][lane * 4 + 3] = vx.w;
    } else {
      s_agg[wv][r][lane * 4 + 0] = 0.f; s_agg[wv][r][lane * 4 + 1] = 0.f;
      s_agg[wv][r][lane * 4 + 2] = 0.f; s_agg[wv][r][lane * 4 + 3] = 0.f;
      s_in[wv][r][lane * 4 + 0] = 0.f;  s_in[wv][r][lane * 4 + 1] = 0.f;
      s_in[wv][r][lane * 4 + 2] = 0.f;  s_in[wv][r][lane * 4 + 3] = 0.f;
    }
  }
  __syncthreads();

  const int m     = lane & 15;         // A-frag row / B-frag & C-frag column
  const int khalf = (lane >> 4) << 1;  // K sub-slice: 0 or 2
  const int mbase = (lane >> 4) << 3;  // C-frag row base: 0 or 8

  for (int ct = 0; ct < 8; ++ct) {
    const int n0 = ct * 16;
    const float bv = bias[n0 + m];     // bias depends only on N = lane&15
    v8f c;
    #pragma unroll
    for (int v = 0; v < 8; ++v) c[v] = bv;

    const float* wlrow = Wl + (long long)(n0 + m) * NODE_DIM;
    const float* wrrow = Wr + (long long)(n0 + m) * NODE_DIM;

    #pragma unroll 4
    for (int k0 = 0; k0 < NODE_DIM; k0 += 4) {
      v2f aA, aX;
      aA[0] = s_agg[wv][m][k0 + khalf];
      aA[1] = s_agg[wv][m][k0 + khalf + 1];
      const v2f bL = *reinterpret_cast<const v2f*>(wlrow + k0 + khalf);
      c = __builtin_amdgcn_wmma_f32_16x16x4_f32(false, aA, false, bL,
                                                (short)0, c, false, false);
      aX[0] = s_in[wv][m][k0 + khalf];
      aX[1] = s_in[wv][m][k0 + khalf + 1];
      const v2f bR = *reinterpret_cast<const v2f*>(wrrow + k0 + khalf);
      c = __builtin_amdgcn_wmma_f32_16x16x4_f32(false, aX, false, bR,
                                                (short)0, c, false, false);
    }

    #pragma unroll
    for (int v = 0; v < 8; ++v) s_out[wv][mbase + v][n0 + m] = c[v];
  }
  __syncthreads();

  if (NORM_RELU) {
    // Row-wise L2 norm: lane pair (r, r+16) split the 128 cols of row r.
    const int row = lane & 15;
    const int cb  = (lane >> 4) * 64;
    float ss = 0.f;
    #pragma unroll 8
    for (int c2 = 0; c2 < 64; ++c2) {
      const float v = s_out[wv][row][cb + c2];
      ss += v * v;
    }
    ss += __shfl_xor(ss, 16, 32);
    const float scl = 1.0f / fmaxf(sqrtf(ss), 1e-12f);
    #pragma unroll 8
    for (int c2 = 0; c2 < 64; ++c2) {
      const float v = s_out[wv][row][cb + c2] * scl;
      s_out[wv][row][cb + c2] = fmaxf(v, 0.0f);   // ReLU
    }
    __syncthreads();
  }

  // ---- Coalesced float4 store of the 16x128 tile ----
  #pragma unroll 4
  for (int r = 0; r < 16; ++r) {
    const int n = node0 + r;
    if (n < N) {
      float4 v;
      v.x = s_out[wv][r][lane * 4 + 0];
      v.y = s_out[wv][r][lane * 4 + 1];
      v.z = s_out[wv][r][lane * 4 + 2];
      v.w = s_out[wv][r][lane * 4 + 3];
      *reinterpret_cast<float4*>(outp + (long long)n * NODE_DIM + lane * 4) = v;
    }
  }
}

// ---------------------------------------------------------------------------
extern "C" void kernel_launch(void* const* d_in, const int* in_sizes, int n_in,
                              void* d_out, int out_size, void* d_ws,
                              size_t ws_size, hipStream_t stream) {
  (void)n_in; (void)out_size; (void)ws_size;

  const float* x   = (const float*)d_in[0];
  const int*   ei  = (const int*)d_in[1];
  const float* W1l = (const float*)d_in[2];
  const float* b1  = (const float*)d_in[3];
  const float* W1r = (const float*)d_in[4];
  const float* W2l = (const float*)d_in[5];
  const float* b2  = (const float*)d_in[6];
  const float* W2r = (const float*)d_in[7];
  float* out = (float*)d_out;

  const int N = in_sizes[0] / NODE_DIM;  // 100000
  const int E = in_sizes[1] / 2;         // 1600000
  const int* src = ei;
  const int* dst = ei + E;

  // Workspace layout: cnt[N] | agg[N*128] | h[N*128]  (~103 MB)
  char* ws = (char*)d_ws;
  float* cnt = (float*)ws;
  size_t off = ((size_t)N * 4 + 255) & ~(size_t)255;
  float* aggb = (float*)(ws + off);
  off += (((size_t)N * NODE_DIM * 4) + 255) & ~(size_t)255;
  float* h = (float*)(ws + off);

  const int tiles  = (N + 15) / 16;
  const int cblk   = (tiles + WAVES_PER_BLOCK - 1) / WAVES_PER_BLOCK;
  const long long sthreads = (long long)E * 32;
  const int sblk = (int)((sthreads + 255) / 256);
  const int dblk = (E + 255) / 256;

  // ---- Layer 1 ----
  hipMemsetAsync(cnt, 0, (size_t)N * 4, stream);
  hipMemsetAsync(aggb, 0, (size_t)N * NODE_DIM * 4, stream);
  sage_count_kernel<<<dblk, 256, 0, stream>>>(dst, cnt, E);
  sage_scatter_kernel<<<sblk, 256, 0, stream>>>(x, src, dst, aggb, E);
  sage_combine_kernel<true><<<cblk, WAVES_PER_BLOCK * 32, 0, stream>>>(
      aggb, cnt, x, W1l, b1, W1r, h, N);

  // ---- Layer 2 ----
  hipMemsetAsync(aggb, 0, (size_t)N * NODE_DIM * 4, stream);
  sage_scatter_kernel<<<sblk, 256, 0, stream>>>(h, src, dst, aggb, E);
  sage_combine_kernel<false><<<cblk, WAVES_PER_BLOCK * 32, 0, stream>>>(
      aggb, cnt, h, W2l, b2, W2r, out, N);
}